// Hd_35399120454206
// MI455X (gfx1250) — compile-verified
//
#include <hip/hip_runtime.h>
#include <math.h>

#define B1N 32
#define B2N 32
#define LN  256
#define FN  64
#define PAD 68          // padded LDS row stride (floats): conflict-free b64 frag loads
#define BIGF 1e9f

typedef __attribute__((ext_vector_type(2))) float v2f;
typedef __attribute__((ext_vector_type(8))) float v8f;

// One block per (a,b) batch pair. 256 threads = 8 waves (wave32).
// Wave w owns i-tiles {w, w+8}; loops all 16 j-tiles.
// Core: V_WMMA_F32_16X16X4_F32, K stepped 16x (two independent accumulator
// chains) to cover F=64 in full f32 precision.
// All reductions happen on squared distances (sqrt once at the end);
// validity masks are folded into the norms (BIG poisoning), so the inner
// loop has zero per-element selects and zero sqrts.
__global__ __launch_bounds__(256) void hausdorff_wmma_kernel(
    const float* __restrict__ v1, const int* __restrict__ sz1,
    const float* __restrict__ v2, const int* __restrict__ sz2,
    float* __restrict__ out)
{
    extern __shared__ __align__(16) float smem[];
    float* s2     = smem;               // [LN][PAD] staged v2[b]
    float* n1s    = s2 + LN * PAD;      // [LN] ||v1_i||^2 (+BIG if i invalid)
    float* n2s    = n1s + LN;           // [LN] ||v2_j||^2 (+BIG if j invalid)
    int*   colmin = (int*)(n2s + LN);   // [LN] per-column min d2, float bits (>=0)
    int*   obits  = colmin + LN;        // final max accumulator, float bits

    const int b    = blockIdx.x;
    const int a    = blockIdx.y;
    const int tid  = threadIdx.x;
    const int lane = tid & 31;
    const int wave = tid >> 5;          // 0..7
    const int half = lane >> 4;         // 0 or 1 (lane group)
    const int l16  = lane & 15;

    const int s1a = sz1[a];
    const int s2b = sz2[b];

    const float* v1base = v1 + (size_t)a * LN * FN;
    const float* v2base = v2 + (size_t)b * LN * FN;

    // ---- stage v2[b] into padded LDS (coalesced float4 copy) ----
    #pragma unroll
    for (int e = 0; e < (LN * FN) / (256 * 4); ++e) {   // 16 iterations
        int gidx = (e * 256 + tid) * 4;                 // flat float index
        int row  = gidx >> 6;                           // /FN
        int col  = gidx & 63;
        float4 val = *(const float4*)(v2base + gidx);
        *(float4*)(s2 + row * PAD + col) = val;
    }
    if (tid < LN) colmin[tid] = __float_as_int(BIGF);
    if (tid == 0) *obits = 0;                           // bits of 0.0f
    __syncthreads();

    // ---- squared norms (+BIG poison for invalid rows/cols) ----
    {
        const float* r2 = s2 + tid * PAD;
        float acc2 = 0.f;
        #pragma unroll
        for (int k = 0; k < FN; k += 4) {
            float4 x = *(const float4*)(r2 + k);
            acc2 += x.x * x.x + x.y * x.y + x.z * x.z + x.w * x.w;
        }
        n2s[tid] = acc2 + ((tid < s2b) ? 0.0f : BIGF);
        const float* r1 = v1base + tid * FN;
        float acc1 = 0.f;
        #pragma unroll
        for (int k = 0; k < FN; k += 4) {
            float4 x = *(const float4*)(r1 + k);
            acc1 += x.x * x.x + x.y * x.y + x.z * x.z + x.w * x.w;
        }
        n1s[tid] = acc1 + ((tid < s1a) ? 0.0f : BIGF);
    }
    __syncthreads();

    // ---- main compute: per wave, 2 i-tiles x 16 j-tiles of 16x16 ----
    for (int itsel = 0; itsel < 2; ++itsel) {
        const int it   = wave + itsel * 8;          // i-tile index 0..15
        const int irow = it * 16 + half * 8;        // this lane-group's M base

        // A fragments (16x4 f32): lane = M (l16), (half,elem) = K within 4-chunk
        v2f afrag[16];
        const float* arow = v1base + (size_t)(it * 16 + l16) * FN + half * 2;
        #pragma unroll
        for (int kk = 0; kk < 16; ++kk)
            afrag[kk] = *(const v2f*)(arow + kk * 4);

        float n1v[8];
        #pragma unroll
        for (int r = 0; r < 8; ++r) n1v[r] = n1s[irow + r];

        float rowpart[8];                           // per-lane min over j of d2
        #pragma unroll
        for (int r = 0; r < 8; ++r) rowpart[r] = BIGF;

        for (int jt = 0; jt < 16; ++jt) {
            v8f acc0 = {};
            v8f acc1 = {};
            // B fragments (4x16 f32): lane = N (l16), (half,elem) = K; B[k][n]=v2[n][k]
            const float* brow = s2 + (jt * 16 + l16) * PAD + half * 2;
            #pragma unroll
            for (int kk = 0; kk < 16; kk += 2) {
                v2f b0 = *(const v2f*)(brow + kk * 4);
                v2f b1 = *(const v2f*)(brow + (kk + 1) * 4);
                acc0 = __builtin_amdgcn_wmma_f32_16x16x4_f32(
                    false, afrag[kk], false, b0, (short)0, acc0, false, false);
                acc1 = __builtin_amdgcn_wmma_f32_16x16x4_f32(
                    false, afrag[kk + 1], false, b1, (short)0, acc1, false, false);
            }

            const int   jcol = jt * 16 + l16;
            const float n2v  = n2s[jcol];
            float cmin = BIGF;
            #pragma unroll
            for (int r = 0; r < 8; ++r) {
                float dot = acc0[r] + acc1[r];
                float d2  = n1v[r] + n2v - 2.0f * dot;  // BIG-poisoned if i or j invalid
                rowpart[r] = fminf(rowpart[r], d2);
                cmin       = fminf(cmin, d2);
            }
            cmin = fminf(cmin, __shfl_xor(cmin, 16, 32));
            cmin = fmaxf(cmin, 0.0f);                   // keep int-bit ordering valid
            if (half == 0)
                atomicMin(&colmin[jcol], __float_as_int(cmin));
        }

        // row direction: reduce rowpart across the 16 lanes of the group (once
        // per i-tile), then masked max over this wave's valid rows.
        float m1p = 0.0f;                               // neutral: d2 >= 0 after clamp
        #pragma unroll
        for (int r = 0; r < 8; ++r) {
            float rm = rowpart[r];
            rm = fminf(rm, __shfl_xor(rm, 1, 32));
            rm = fminf(rm, __shfl_xor(rm, 2, 32));
            rm = fminf(rm, __shfl_xor(rm, 4, 32));
            rm = fminf(rm, __shfl_xor(rm, 8, 32));
            const int i = irow + r;
            if (i < s1a) m1p = fmaxf(m1p, rm);
        }
        m1p = fmaxf(m1p, __shfl_xor(m1p, 16, 32));
        m1p = fmaxf(m1p, 0.0f);
        if (lane == 0) atomicMax(obits, __float_as_int(m1p));
    }
    __syncthreads();

    // column direction: max over valid columns of colmin
    {
        float cv = (tid < s2b) ? __int_as_float(colmin[tid]) : 0.0f;
        cv = fmaxf(cv, __shfl_xor(cv, 1, 32));
        cv = fmaxf(cv, __shfl_xor(cv, 2, 32));
        cv = fmaxf(cv, __shfl_xor(cv, 4, 32));
        cv = fmaxf(cv, __shfl_xor(cv, 8, 32));
        cv = fmaxf(cv, __shfl_xor(cv, 16, 32));
        if (lane == 0) atomicMax(obits, __float_as_int(cv));
    }
    __syncthreads();

    // single sqrt per (a,b) pair; clamp commutes with min/max
    if (tid == 0)
        out[(size_t)a * B2N + b] = sqrtf(fmaxf(__int_as_float(*obits), 0.0f));
}

extern "C" void kernel_launch(void* const* d_in, const int* in_sizes, int n_in,
                              void* d_out, int out_size, void* d_ws, size_t ws_size,
                              hipStream_t stream) {
    const float* v1  = (const float*)d_in[0];
    const int*   sz1 = (const int*)d_in[1];
    const float* v2  = (const float*)d_in[2];
    const int*   sz2 = (const int*)d_in[3];
    float* out = (float*)d_out;

    const size_t shmem = (size_t)(LN * PAD + 2 * LN) * sizeof(float)
                       + (size_t)(LN + 1) * sizeof(int);   // ~71 KB
    (void)hipFuncSetAttribute(reinterpret_cast<const void*>(hausdorff_wmma_kernel),
                              hipFuncAttributeMaxDynamicSharedMemorySize, (int)shmem);

    dim3 grid(B2N, B1N);
    hausdorff_wmma_kernel<<<grid, 256, shmem, stream>>>(v1, sz1, v2, sz2, out);
}